// XFeat_33904471835513
// MI455X (gfx1250) — compile-verified
//
#include <hip/hip_runtime.h>
#include <math.h>

// ---------------- problem constants ----------------
#define BB    8
#define HF    128
#define WF    128
#define HH    1024
#define WW    1024
#define NPIX  (HH*WW)          // 1048576 pixels per batch
#define CCH   64
#define TOPK  4096
#define NMS_THR 0.05f
#define TSTRIDE 20             // LDS tile row stride (floats): 80 B, 16-B aligned

typedef float v2f __attribute__((ext_vector_type(2)));
typedef float v8f __attribute__((ext_vector_type(8)));

// ---------------- workspace layout (bytes) ----------------
#define WS_HM      ((size_t)0)
#define WS_ROWMAX  (WS_HM     + (size_t)BB*NPIX*4)          // 33.5 MB
#define WS_KEYS    (WS_ROWMAX + (size_t)BB*NPIX*4)          // 33.5 MB
#define WS_M1T     (WS_KEYS   + (size_t)BB*NPIX*4)          // 33.5 MB
#define WS_LISTA   (WS_M1T    + (size_t)BB*CCH*HF*WF*4)     // 33.5 MB
#define WS_LISTB   (WS_LISTA  + (size_t)BB*TOPK*8)
#define WS_BLKOFF  (WS_LISTB  + (size_t)BB*TOPK*4)
#define WS_HIST    (WS_BLKOFF + (size_t)BB*4096*4)
#define WS_STATE   (WS_HIST   + (size_t)4*BB*256*4)
#define WS_CTRA    (WS_STATE  + (size_t)BB*16)

// order-preserving float -> uint (bigger key == bigger float)
__device__ __forceinline__ unsigned enc(float f) {
  unsigned u = __float_as_uint(f);
  return (u & 0x80000000u) ? ~u : (u | 0x80000000u);
}
__device__ __forceinline__ float dec(unsigned k) {
  unsigned u = (k & 0x80000000u) ? (k & 0x7FFFFFFFu) : ~k;
  return __uint_as_float(u);
}

// LDS byte offset of a __shared__ object: flat-aperture rule -> low 32 bits
__device__ __forceinline__ unsigned lds_off(const void* p) {
  return (unsigned)(unsigned long long)(size_t)p;
}

// torch bicubic kernel, a = -0.75
__device__ __forceinline__ float cubw(float x) {
  float ax = fabsf(x), ax2 = ax*ax, ax3 = ax2*ax;
  const float a = -0.75f;
  float wn = (a+2.0f)*ax3 - (a+3.0f)*ax2 + 1.0f;
  float wf = a*ax3 - 5.0f*a*ax2 + 8.0f*a*ax - 4.0f*a;
  return (ax <= 1.0f) ? wn : ((ax < 2.0f) ? wf : 0.0f);
}

// ---------------- init: zero histograms / counters, seed radix state --------
__global__ void k_init(unsigned* __restrict__ hist, uint4* __restrict__ state,
                       unsigned* __restrict__ ctrA) {
  int i = blockIdx.x*256 + threadIdx.x;
  if (i < 4*BB*256) hist[i] = 0u;
  if (i < BB) { ctrA[i] = 0u; state[i] = make_uint4(0u, 0u, 0u, (unsigned)TOPK); }
}

// ---------------- softmax over 65 logits + 8x8 pixel shuffle ----------------
__global__ void k_softmax_shuffle(const float* __restrict__ K1, float* __restrict__ hm) {
  int b = blockIdx.y;
  int cell = blockIdx.x*256 + threadIdx.x;      // 0..16383
  int yf = cell >> 7, xf = cell & 127;
  const float* p = K1 + (size_t)b*65*HF*WF + cell;
  float v[65];
  float m = -3.402823466e38f;
#pragma unroll
  for (int c = 0; c < 65; ++c) { v[c] = p[(size_t)c*HF*WF]; m = fmaxf(m, v[c]); }
  float s = 0.f;
#pragma unroll
  for (int c = 0; c < 65; ++c) { v[c] = expf(v[c] - m); s += v[c]; }
  float inv = 1.0f / s;
  float* o = hm + (size_t)b*NPIX;
#pragma unroll
  for (int c = 0; c < 64; ++c) {
    int dy = c >> 3, dx = c & 7;
    o[(size_t)(yf*8 + dy)*WW + (xf*8 + dx)] = v[c]*inv;
  }
}

// ---------------- separable 5x5 NMS: horizontal pass ----------------
__global__ void k_rowmax(const float* __restrict__ hm, float* __restrict__ rmax) {
  int b = blockIdx.y;
  int pix = blockIdx.x*256 + threadIdx.x;
  int y = pix >> 10, x = pix & 1023;
  const float* r = hm + (size_t)b*NPIX + (size_t)y*WW;
  float m = -3.402823466e38f;
#pragma unroll
  for (int d = -2; d <= 2; ++d) { int xx = x + d; if (xx >= 0 && xx < WW) m = fmaxf(m, r[xx]); }
  rmax[(size_t)b*NPIX + pix] = m;
}

// ---- vertical NMS pass + nearest(heatmap)*bilinear(H1) score -> ordered key
__global__ void k_scores(const float* __restrict__ hm, const float* __restrict__ rmax,
                         const float* __restrict__ H1, unsigned* __restrict__ keys) {
  int b = blockIdx.y;
  int pix = blockIdx.x*256 + threadIdx.x;
  int y = pix >> 10, x = pix & 1023;
  const float* hb = hm + (size_t)b*NPIX;
  const float* rb = rmax + (size_t)b*NPIX;
  float lm = -3.402823466e38f;
#pragma unroll
  for (int d = -2; d <= 2; ++d) { int yy = y + d; if (yy >= 0 && yy < HH) lm = fmaxf(lm, rb[(size_t)yy*WW + x]); }
  float v = hb[pix];
  bool mask = (v == lm) && (v > NMS_THR);
  float score = -1.0f;
  if (mask) {
    float gx = 2.0f*(float)x/1023.0f - 1.0f;
    float gy = 2.0f*(float)y/1023.0f - 1.0f;
    // nearest sample of full-res heatmap (align_corners=False unnorm, round-half-even)
    float fx = ((gx + 1.0f)*1024.0f - 1.0f)*0.5f;
    float fy = ((gy + 1.0f)*1024.0f - 1.0f)*0.5f;
    int ix = (int)rintf(fx), iy = (int)rintf(fy);
    float sn = (ix >= 0 && ix < WW && iy >= 0 && iy < HH) ? hb[(size_t)iy*WW + ix] : 0.0f;
    // bilinear sample of 128x128 reliability map, zero padding
    float px = ((gx + 1.0f)*128.0f - 1.0f)*0.5f;
    float py = ((gy + 1.0f)*128.0f - 1.0f)*0.5f;
    float x0f = floorf(px), y0f = floorf(py);
    float tx = px - x0f, ty = py - y0f;
    int x0 = (int)x0f, y0 = (int)y0f;
    const float* Hb = H1 + (size_t)b*HF*WF;
    float sb = 0.f;
#pragma unroll
    for (int dy = 0; dy < 2; ++dy) {
      int yy = y0 + dy; float wy = dy ? ty : (1.f - ty);
#pragma unroll
      for (int dx = 0; dx < 2; ++dx) {
        int xx = x0 + dx; float wx = dx ? tx : (1.f - tx);
        if (xx >= 0 && xx < WF && yy >= 0 && yy < HF) sb += Hb[yy*WF + xx]*(wx*wy);
      }
    }
    score = sn*sb;
  }
  keys[(size_t)b*NPIX + pix] = enc(score);
}

// ---------------- radix-select: per-batch 256-bin histogram of byte `pass` --
__global__ void k_hist(const unsigned* __restrict__ keys, const uint4* __restrict__ state,
                       unsigned* __restrict__ hist, int pass) {
  __shared__ unsigned hloc[256];
  int b = blockIdx.y;
  hloc[threadIdx.x] = 0u;
  __syncthreads();
  uint4 st = state[b];                       // {prefix, mask, countAbove, remaining}
  int pix = blockIdx.x*256 + threadIdx.x;
  const unsigned* kp = keys + (size_t)b*NPIX + pix;
  __builtin_prefetch(kp + 256, 0, 0);        // global_prefetch_b8
  unsigned k = *kp;
  if ((k & st.y) == st.x) {
    unsigned byte = (k >> (24 - 8*pass)) & 255u;
    atomicAdd(&hloc[byte], 1u);
  }
  __syncthreads();
  unsigned c = hloc[threadIdx.x];
  if (c) atomicAdd(&hist[((size_t)pass*BB + b)*256 + threadIdx.x], c);
}

// ---------------- radix-select: pick the bin holding the K-th largest -------
__global__ void k_pick(const unsigned* __restrict__ hist, uint4* __restrict__ state, int pass) {
  int b = blockIdx.x;
  if (threadIdx.x == 0) {
    uint4 st = state[b];
    const unsigned* hb = &hist[((size_t)pass*BB + b)*256];
    unsigned rem = st.w;
    unsigned cum = 0; int sel = 0;
    for (int bin = 255; bin >= 0; --bin) {
      unsigned h = hb[bin];
      if (cum + h >= rem) { sel = bin; break; }
      cum += h;
    }
    int shift = 24 - 8*pass;
    st.x |= ((unsigned)sel) << shift;
    st.y |= 255u << shift;
    st.z += cum;            // elements strictly greater so far
    st.w  = rem - cum;      // still needed from within selected bin
    state[b] = st;
  }
}

// --- compact strictly-greater elems into listA; count key==T ties per block -
__global__ void k_compact(const unsigned* __restrict__ keys, const uint4* __restrict__ state,
                          uint2* __restrict__ listA, unsigned* __restrict__ ctrA,
                          unsigned* __restrict__ blkcnt) {
  __shared__ unsigned woff[8];
  int b = blockIdx.y;
  unsigned T = state[b].x;
  int pix = blockIdx.x*256 + threadIdx.x;
  unsigned k = keys[(size_t)b*NPIX + pix];
  if (k > T) {
    unsigned pos = atomicAdd(&ctrA[b], 1u);   // guaranteed < TOPK
    if (pos < TOPK) listA[(size_t)b*TOPK + pos] = make_uint2(k, (unsigned)pix);
  }
  bool tie = (k == T);
  unsigned long long m = __ballot(tie);
  int lane = threadIdx.x & 31, wave = threadIdx.x >> 5;
  if (lane == 0) woff[wave] = (unsigned)__popcll(m);
  __syncthreads();
  if (threadIdx.x == 0) {
    unsigned s = 0;
    for (int w = 0; w < 8; ++w) s += woff[w];
    blkcnt[(size_t)b*4096 + blockIdx.x] = s;
  }
}

// ---------------- exclusive scan of 4096 per-block tie counts ---------------
__global__ void k_scan(unsigned* __restrict__ blk) {
  __shared__ unsigned s[1024];
  int b = blockIdx.x, t = threadIdx.x;
  unsigned* base = blk + (size_t)b*4096;
  unsigned v0 = base[4*t], v1 = base[4*t+1], v2 = base[4*t+2], v3 = base[4*t+3];
  unsigned sum = v0 + v1 + v2 + v3;
  s[t] = sum;
  for (unsigned off = 1; off < 1024; off <<= 1) {
    __syncthreads();
    unsigned x = (t >= (int)off) ? s[t - off] : 0u;
    __syncthreads();
    s[t] += x;
  }
  __syncthreads();
  unsigned e = t ? s[t - 1] : 0u;
  base[4*t]   = e;
  base[4*t+1] = e + v0;
  base[4*t+2] = e + v0 + v1;
  base[4*t+3] = e + v0 + v1 + v2;
}

// ----- emit lowest-index ties (key==T) into listB at their global rank ------
__global__ void k_emit(const unsigned* __restrict__ keys, const uint4* __restrict__ state,
                       const unsigned* __restrict__ blkoff, unsigned* __restrict__ listB) {
  __shared__ unsigned woff[8];
  int b = blockIdx.y;
  uint4 st = state[b];
  unsigned T = st.x, needed = st.w;
  int pix = blockIdx.x*256 + threadIdx.x;
  unsigned k = keys[(size_t)b*NPIX + pix];
  bool tie = (k == T);
  unsigned long long m = __ballot(tie);
  int lane = threadIdx.x & 31, wave = threadIdx.x >> 5;
  if (lane == 0) woff[wave] = (unsigned)__popcll(m);
  __syncthreads();
  if (threadIdx.x == 0) {
    unsigned s = 0;
    for (int w = 0; w < 8; ++w) { unsigned c = woff[w]; woff[w] = s; s += c; }
  }
  __syncthreads();
  if (tie) {
    unsigned r = blkoff[(size_t)b*4096 + blockIdx.x] + woff[wave]
               + (unsigned)__popcll(m & ((1ull << lane) - 1ull));
    if (r < needed) listB[(size_t)b*TOPK + r] = (unsigned)pix;
  }
}

// --------- bitonic sort of listA: (key desc, idx asc) via 64-bit combo ------
__global__ void k_sort(uint2* __restrict__ listA, const unsigned* __restrict__ ctrA) {
  __shared__ unsigned long long s[TOPK];     // 32 KB LDS
  int b = blockIdx.x, t = threadIdx.x;
  unsigned cnt = ctrA[b]; if (cnt > TOPK) cnt = TOPK;
  uint2* L = listA + (size_t)b*TOPK;
  for (int q = 0; q < 4; ++q) {
    int j = t + q*1024;
    unsigned long long v = 0ull;              // padding sorts last
    if ((unsigned)j < cnt) { uint2 u = L[j]; v = ((unsigned long long)u.x << 32) | (unsigned)(~u.y); }
    s[j] = v;
  }
  for (unsigned size = 2; size <= TOPK; size <<= 1) {
    for (unsigned stride = size >> 1; stride > 0; stride >>= 1) {
      __syncthreads();
#pragma unroll
      for (int q = 0; q < 2; ++q) {
        unsigned p = (unsigned)t + (unsigned)q*1024u;
        unsigned i = 2u*p - (p & (stride - 1u));
        unsigned j = i + stride;
        bool desc = ((i & size) == 0u);
        unsigned long long a = s[i], c = s[j];
        bool sw = desc ? (a < c) : (a > c);
        if (sw) { s[i] = c; s[j] = a; }
      }
    }
  }
  __syncthreads();
  for (int q = 0; q < 4; ++q) {
    int j = t + q*1024;
    unsigned long long v = s[j];
    L[j] = make_uint2((unsigned)(v >> 32), ~(unsigned)v);
  }
}

// ---------------- write pts = (x, y, score) ----------------
__global__ void k_pts(const uint2* __restrict__ listA, const unsigned* __restrict__ listB,
                      const uint4* __restrict__ state, float* __restrict__ pts) {
  int b = blockIdx.y;
  int k = blockIdx.x*256 + threadIdx.x;
  uint4 st = state[b];
  unsigned T = st.x, cgt = st.z;
  unsigned key, idx;
  if ((unsigned)k < cgt) { uint2 u = listA[(size_t)b*TOPK + k]; key = u.x; idx = u.y; }
  else                   { idx = listB[(size_t)b*TOPK + ((unsigned)k - cgt)]; key = T; }
  float score = dec(key);
  float x = (float)(idx & 1023u), y = (float)(idx >> 10);
  float* o = pts + ((size_t)b*TOPK + k)*3;
  o[0] = x; o[1] = y; o[2] = score;
}

// ---- channel-L2-normalize M1 and transpose to (b,y,x,c) --------------------
// One wave per tile of 16 spatial positions x 64 channels.
//  * tile load:  GLOBAL_LOAD_ASYNC_TO_LDS_B128 (ASYNCcnt path, no VGPR trip)
//  * sumsq:      V_WMMA_F32_16X16X4_F32 with B = ones (column-sum reduction)
//  * store:      normalize into contiguous LDS stage, then
//                GLOBAL_STORE_ASYNC_FROM_LDS_B128
__global__ void k_normT(const float* __restrict__ M1, float* __restrict__ M1t) {
  __shared__ __align__(16) float tile[64*TSTRIDE];  // [ch][pos], 80 B stride
  __shared__ __align__(16) float stage[16*CCH];     // [pos][ch] output layout
  __shared__ float invs[16];
  int b = blockIdx.y;
  int posbase = blockIdx.x * 16;
  int lane = threadIdx.x;          // block = 32 threads = 1 wave, EXEC all ones
  __builtin_amdgcn_s_wait_tensorcnt(0);
  const float* src = M1 + (size_t)b*CCH*HF*WF + posbase;

  // ---- async load: 256 x B128 transfers cover the 64x16 f32 tile ----
  unsigned tbase = lds_off(tile);
#pragma unroll
  for (int it = 0; it < 8; ++it) {
    int t = it*32 + lane;
    int ch = t >> 2, pos4 = (t & 3) << 2;          // 4 consecutive positions
    unsigned loff = tbase + (unsigned)(ch*TSTRIDE + pos4)*4u;
    const float* g = src + (size_t)ch*HF*WF + pos4;
    asm volatile("global_load_async_to_lds_b128 %0, %1, off"
                 :: "v"(loff), "v"((unsigned long long)(size_t)g) : "memory");
  }
  asm volatile("s_wait_asynccnt 0x0" ::: "memory");
  __syncthreads();

  // ---- sumsq over channels via WMMA ones-reduction ----
  v8f acc = {};
  v2f bones = {1.0f, 1.0f};
#pragma unroll
  for (int kc = 0; kc < 16; ++kc) {
    // A 16x4 layout: lanes 0-15 -> M=lane,K={0,1}; lanes 16-31 -> M=lane-16,K={2,3}
    int ch0 = 4*kc + ((lane >> 4) << 1);
    int pos = lane & 15;
    float a0 = tile[ch0*TSTRIDE + pos];
    float a1 = tile[(ch0 + 1)*TSTRIDE + pos];
    v2f a; a[0] = a0*a0; a[1] = a1*a1;
    acc = __builtin_amdgcn_wmma_f32_16x16x4_f32(false, a, false, bones,
                                                (short)0, acc, false, false);
  }
  // C/D layout: VGPR r -> M=r (lanes 0-15) / M=8+r (lanes 16-31); all N equal
  if ((lane & 15) == 0) {
    int mb = (lane >> 4)*8;
#pragma unroll
    for (int r = 0; r < 8; ++r) {
      float ss = acc[r];
      invs[mb + r] = 1.0f / fmaxf(sqrtf(ss), 1e-12f);
    }
  }
  __syncthreads();

  // ---- normalize + transpose into contiguous stage ----
#pragma unroll
  for (int c2 = 0; c2 < 32; ++c2) {
    int pos = c2 >> 1;
    int ch = ((c2 & 1) << 5) + lane;
    stage[pos*CCH + ch] = tile[ch*TSTRIDE + pos]*invs[pos];
  }
  __syncthreads();
  asm volatile("s_wait_dscnt 0x0" ::: "memory");   // stage writes visible to async

  // ---- async store: 256 x B128 transfers, coalesced 256B per position ----
  unsigned sbase = lds_off(stage);
  float* dst = M1t + ((size_t)b*HF*WF + posbase)*CCH;
#pragma unroll
  for (int it = 0; it < 8; ++it) {
    int t = it*32 + lane;
    int pos = t >> 4, ch4 = (t & 15) << 2;         // 4 consecutive channels
    unsigned loff = sbase + (unsigned)(pos*CCH + ch4)*4u;
    float* g = dst + (size_t)pos*CCH + ch4;
    asm volatile("global_store_async_from_lds_b128 %0, %1, off"
                 :: "v"((unsigned long long)(size_t)g), "v"(loff) : "memory");
  }
  asm volatile("s_wait_asynccnt 0x0" ::: "memory");
}

// ---- bicubic sample of normalized/transposed features + final L2 norm ------
// One wave per keypoint; lane handles channels 2*lane, 2*lane+1 (float2 reads)
__global__ void k_feats(const float* __restrict__ M1t, const float* __restrict__ pts,
                        float* __restrict__ feats) {
  int b = blockIdx.y;
  int wave = threadIdx.x >> 5, lane = threadIdx.x & 31;
  int k = blockIdx.x*8 + wave;
  const float* p = pts + ((size_t)b*TOPK + k)*3;
  float px = p[0], py = p[1];
  float gx = 2.0f*px/1023.0f - 1.0f;
  float gy = 2.0f*py/1023.0f - 1.0f;
  float ix = ((gx + 1.0f)*128.0f - 1.0f)*0.5f;
  float iy = ((gy + 1.0f)*128.0f - 1.0f)*0.5f;
  float x0f = floorf(ix), y0f = floorf(iy);
  float tx = ix - x0f, ty = iy - y0f;
  int x0 = (int)x0f, y0 = (int)y0f;
  float wx[4] = {cubw(tx + 1.0f), cubw(tx), cubw(1.0f - tx), cubw(2.0f - tx)};
  float wy[4] = {cubw(ty + 1.0f), cubw(ty), cubw(1.0f - ty), cubw(2.0f - ty)};
  const float2* Mb = (const float2*)(M1t + (size_t)b*HF*WF*CCH);
  float a0 = 0.f, a1 = 0.f;
#pragma unroll
  for (int j = 0; j < 4; ++j) {
    int yt = y0 + j - 1;
    if (yt < 0 || yt >= HF) continue;          // uniform across the wave
#pragma unroll
    for (int i = 0; i < 4; ++i) {
      int xt = x0 + i - 1;
      if (xt < 0 || xt >= WF) continue;
      float w = wy[j]*wx[i];
      float2 v = Mb[(size_t)(yt*WF + xt)*32 + lane];
      a0 += w*v.x; a1 += w*v.y;
    }
  }
  float ss = a0*a0 + a1*a1;
#pragma unroll
  for (int off = 16; off > 0; off >>= 1) ss += __shfl_xor(ss, off, 32);
  float inv = 1.0f / fmaxf(sqrtf(ss), 1e-12f);
  float2* out = (float2*)(feats + ((size_t)b*TOPK + k)*CCH);
  out[lane] = make_float2(a0*inv, a1*inv);
}

// ---------------- launch ----------------
extern "C" void kernel_launch(void* const* d_in, const int* in_sizes, int n_in,
                              void* d_out, int out_size, void* d_ws, size_t ws_size,
                              hipStream_t stream) {
  const float* M1 = (const float*)d_in[0];   // (8,64,128,128)
  const float* K1 = (const float*)d_in[1];   // (8,65,128,128)
  const float* H1 = (const float*)d_in[2];   // (8,1,128,128)
  char* ws = (char*)d_ws;
  float*    hm    = (float*)   (ws + WS_HM);
  float*    rmax  = (float*)   (ws + WS_ROWMAX);
  unsigned* keys  = (unsigned*)(ws + WS_KEYS);
  float*    M1t   = (float*)   (ws + WS_M1T);
  uint2*    listA = (uint2*)   (ws + WS_LISTA);
  unsigned* listB = (unsigned*)(ws + WS_LISTB);
  unsigned* blk   = (unsigned*)(ws + WS_BLKOFF);
  unsigned* hist  = (unsigned*)(ws + WS_HIST);
  uint4*    state = (uint4*)   (ws + WS_STATE);
  unsigned* ctrA  = (unsigned*)(ws + WS_CTRA);
  float* pts   = (float*)d_out;                    // (8,4096,3)
  float* feats = pts + (size_t)BB*TOPK*3;          // (8,4096,64)

  k_init<<<dim3(32), dim3(256), 0, stream>>>(hist, state, ctrA);
  k_softmax_shuffle<<<dim3(64, BB), dim3(256), 0, stream>>>(K1, hm);
  k_normT<<<dim3(1024, BB), dim3(32), 0, stream>>>(M1, M1t);
  k_rowmax<<<dim3(4096, BB), dim3(256), 0, stream>>>(hm, rmax);
  k_scores<<<dim3(4096, BB), dim3(256), 0, stream>>>(hm, rmax, H1, keys);
  for (int p = 0; p < 4; ++p) {
    k_hist<<<dim3(4096, BB), dim3(256), 0, stream>>>(keys, state, hist, p);
    k_pick<<<dim3(BB), dim3(32), 0, stream>>>(hist, state, p);
  }
  k_compact<<<dim3(4096, BB), dim3(256), 0, stream>>>(keys, state, listA, ctrA, blk);
  k_scan<<<dim3(BB), dim3(1024), 0, stream>>>(blk);
  k_emit<<<dim3(4096, BB), dim3(256), 0, stream>>>(keys, state, blk, listB);
  k_sort<<<dim3(BB), dim3(1024), 0, stream>>>(listA, ctrA);
  k_pts<<<dim3(16, BB), dim3(256), 0, stream>>>(listA, listB, state, pts);
  k_feats<<<dim3(512, BB), dim3(256), 0, stream>>>(M1t, pts, feats);
}